// DCConv3dKernelPolynomials_71382356459674
// MI455X (gfx1250) — compile-verified
//
#include <hip/hip_runtime.h>
#include <math.h>

// D = A x B + C with V_WMMA_F32_16X16X4_F32 (fp32).
typedef __attribute__((ext_vector_type(2))) float v2f;
typedef __attribute__((ext_vector_type(8))) float v8f;

#define XN_TOTAL   110592   // 4096 * 27 output grid points
#define NB         30       // basis count, padded to 32 for K
#define LDS_STRIDE 34       // row stride in dwords (even -> b64 aligned, odd-ish vs 64 banks)
#define PTS_BLK    256      // points per block (one per thread)

__global__ __launch_bounds__(256)
void dcconv3d_basis_wmma(const float* __restrict__ position,  // (110592, 3) r,theta,phi
                         const float* __restrict__ coeffs,    // (256, 30)  (o*16+i, b)
                         float* __restrict__ out)              // (256, 110592)
{
    __shared__ float sB[PTS_BLK * LDS_STRIDE];  // basis  [point][k], k padded to 32
    __shared__ float sC[256 * LDS_STRIDE];      // coeffs [oi][k],    k padded to 32

    const int tid = threadIdx.x;

    // ---- stage coeffs into LDS (row oi = tid), zero-pad K to 32 ----
    {
        const float* src = coeffs + tid * NB;
        float* dst = sC + tid * LDS_STRIDE;
        #pragma unroll
        for (int k = 0; k < NB; ++k) dst[k] = src[k];
        dst[30] = 0.0f; dst[31] = 0.0f;
    }

    // ---- evaluate 30 hydrogen-wavefunction basis values for this thread's point ----
    {
        const int p = blockIdx.x * PTS_BLK + tid;           // 432*256 == 110592 exactly
        const float r  = position[p * 3 + 0];
        const float th = position[p * 3 + 1];
        const float ph = position[p * 3 + 2];

        // radial parts: R_nl = norm * exp(-r/n) * rho^l * Laguerre_{n-l-1}^{2l+1}(rho), rho=2r/n
        const float e1 = expf(-r);
        const float e2 = expf(-0.5f * r);
        const float e3 = expf(-r * 0.33333333333333f);
        const float e4 = expf(-0.25f * r);
        const float rho2 = r;
        const float rho3 = 0.66666666666667f * r;
        const float rho4 = 0.5f * r;

        const float R10 = 2.0f * e1;
        const float R20 = 0.35355339f * e2 * (2.0f - rho2);
        const float R21 = 0.20412415f * e2 * rho2;
        const float R30 = 0.12830006f * e3 * (3.0f + rho3 * (-3.0f + 0.5f * rho3));
        const float R31 = 0.04536092f * e3 * rho3 * (4.0f - rho3);
        const float R32 = 0.02028534f * e3 * rho3 * rho3;
        const float R40 = 0.0625f     * e4 * (4.0f + rho4 * (-6.0f + rho4 * (2.0f - rho4 * 0.16666666666667f)));
        const float R41 = 0.01613743f * e4 * rho4 * (10.0f + rho4 * (-5.0f + 0.5f * rho4));
        const float R42 = 0.00465847f * e4 * rho4 * rho4 * (6.0f - rho4);
        const float R43 = 0.00176074f * e4 * rho4 * rho4 * rho4;

        // angular parts: real spherical harmonics, Condon-Shortley P_l^m
        const float ct = cosf(th);
        const float st = sqrtf(fmaxf(1.0f - ct * ct, 0.0f));
        const float cp = cosf(ph), sp = sinf(ph);
        const float c2p = cp * cp - sp * sp, s2p = 2.0f * cp * sp;
        const float c3p = c2p * cp - s2p * sp, s3p = s2p * cp + c2p * sp;

        const float st2 = st * st, st3 = st2 * st, ct2 = ct * ct;
        const float P11 = -st;
        const float P21 = -3.0f * ct * st;
        const float P22 = 3.0f * st2;
        const float P31 = -1.5f * st * (5.0f * ct2 - 1.0f);
        const float P32 = 15.0f * ct * st2;
        const float P33 = -15.0f * st3;

        const float Y00  = 0.28209479f;
        const float Y1m1 = 0.48860251f * P11 * sp;
        const float Y10  = 0.48860251f * ct;
        const float Y11  = 0.48860251f * P11 * cp;
        const float Y2m2 = 0.18209140f * P22 * s2p;
        const float Y2m1 = 0.36418281f * P21 * sp;
        const float Y20  = 0.63078313f * 0.5f * (3.0f * ct2 - 1.0f);
        const float Y21  = 0.36418281f * P21 * cp;
        const float Y22  = 0.18209140f * P22 * c2p;
        const float Y3m3 = 0.03933625f * P33 * s3p;
        const float Y3m2 = 0.09635375f * P32 * s2p;
        const float Y3m1 = 0.30469710f * P31 * sp;
        const float Y30  = 0.74635267f * 0.5f * ct * (5.0f * ct2 - 3.0f);
        const float Y31  = 0.30469710f * P31 * cp;
        const float Y32  = 0.09635375f * P32 * c2p;
        const float Y33  = 0.03933625f * P33 * c3p;

        float* dst = sB + tid * LDS_STRIDE;
        dst[0]  = R10 * Y00;
        dst[1]  = R20 * Y00;
        dst[2]  = R21 * Y1m1;  dst[3]  = R21 * Y10;  dst[4]  = R21 * Y11;
        dst[5]  = R30 * Y00;
        dst[6]  = R31 * Y1m1;  dst[7]  = R31 * Y10;  dst[8]  = R31 * Y11;
        dst[9]  = R32 * Y2m2;  dst[10] = R32 * Y2m1; dst[11] = R32 * Y20;
        dst[12] = R32 * Y21;   dst[13] = R32 * Y22;
        dst[14] = R40 * Y00;
        dst[15] = R41 * Y1m1;  dst[16] = R41 * Y10;  dst[17] = R41 * Y11;
        dst[18] = R42 * Y2m2;  dst[19] = R42 * Y2m1; dst[20] = R42 * Y20;
        dst[21] = R42 * Y21;   dst[22] = R42 * Y22;
        dst[23] = R43 * Y3m3;  dst[24] = R43 * Y3m2; dst[25] = R43 * Y3m1;
        dst[26] = R43 * Y30;   dst[27] = R43 * Y31;  dst[28] = R43 * Y32;
        dst[29] = R43 * Y33;
        dst[30] = 0.0f; dst[31] = 0.0f;              // K padding
    }
    __syncthreads();

    // ---- WMMA contraction: out[oi][xn] = sum_k sC[oi][k] * sB[xn][k] ----
    const int lane = tid & 31;
    const int wave = tid >> 5;
    const int lrow = lane & 15;
    const int khi  = (lane >> 4) << 1;   // lanes 16-31 hold K+2,K+3 of each 4-step

    #pragma unroll
    for (int xi = 0; xi < 2; ++xi) {
        const int xt = wave * 2 + xi;                        // xn tile within block (0..15)
        const float* bbase = sB + (xt * 16 + lrow) * LDS_STRIDE + khi;
        const long   col   = (long)blockIdx.x * PTS_BLK + xt * 16 + lrow;

        for (int ot = 0; ot < 16; ++ot) {                    // oi tile (0..15)
            const float* abase = sC + (ot * 16 + lrow) * LDS_STRIDE + khi;

            v8f c = {0.0f, 0.0f, 0.0f, 0.0f, 0.0f, 0.0f, 0.0f, 0.0f};
            #pragma unroll
            for (int ks = 0; ks < 8; ++ks) {                 // K = 32 in steps of 4
                v2f a = *(const v2f*)(abase + ks * 4);       // A: coeffs 16x4 fragment
                v2f b = *(const v2f*)(bbase + ks * 4);       // B: basis^T 4x16 fragment
                c = __builtin_amdgcn_wmma_f32_16x16x4_f32(
                        false, a, false, b, (short)0, c, false, false);
            }

            // C/D layout: VGPR v -> M = v + 8*(lane>=16), N = lane%16
            const int mbase = ot * 16 + ((lane >> 4) << 3);
            #pragma unroll
            for (int v = 0; v < 8; ++v) {
                __builtin_nontemporal_store(c[v], &out[(long)(mbase + v) * XN_TOTAL + col]);
            }
        }
    }
}

extern "C" void kernel_launch(void* const* d_in, const int* in_sizes, int n_in,
                              void* d_out, int out_size, void* d_ws, size_t ws_size,
                              hipStream_t stream) {
    const float* position = (const float*)d_in[0];   // (4096, 27, 3) fp32
    const float* coeffs   = (const float*)d_in[1];   // (16, 16, 30)  fp32
    float* out            = (float*)d_out;           // (16, 16, 4096, 27) fp32

    dim3 grid(XN_TOTAL / PTS_BLK);                   // 432 blocks
    dim3 block(PTS_BLK);                             // 256 threads = 8 waves
    dcconv3d_basis_wmma<<<grid, block, 0, stream>>>(position, coeffs, out);
}